// GAT_19146964205718
// MI455X (gfx1250) — compile-verified
//
#include <hip/hip_runtime.h>
#include <hip/hip_bf16.h>

#define B_ 32
#define N_ 2048
#define H_ 512
#define NCHUNK 16          // n-chunks of 128 rows for the weighted-V reduction
#define MASK_NEG 1e30f

typedef __attribute__((ext_vector_type(2))) float v2f;
typedef __attribute__((ext_vector_type(8))) float v8f;

// ---------------------------------------------------------------------------
// Kernel A: alpha[b,n] = K[b,n,:].wk + Q[b,:].wq + b0  - (1-adj)*MASK_NEG
// One wave per row; lane L owns h = 4*L + 128*j (float4, fully coalesced).
// ---------------------------------------------------------------------------
__global__ void __launch_bounds__(256) alpha_kernel(
    const float* __restrict__ Q, const float* __restrict__ K,
    const int* __restrict__ adj, const float* __restrict__ w_att,
    const float* __restrict__ b_att, float* __restrict__ alpha)
{
  const int b     = blockIdx.y;
  const int chunk = blockIdx.x;
  const int tid   = threadIdx.x;
  const int wave  = tid >> 5;
  const int lane  = tid & 31;

  // preload wk fragments (w_att[H..2H)) for this lane; also compute Q.wq once
  const float4* wkp = reinterpret_cast<const float4*>(w_att + H_);
  const float4* wqp = reinterpret_cast<const float4*>(w_att);
  const float4* qp  = reinterpret_cast<const float4*>(Q + b * H_);
  float4 wk[4];
  float qdot = 0.0f;
  #pragma unroll
  for (int j = 0; j < 4; ++j) {
    const int fi = lane + 32 * j;
    wk[j] = wkp[fi];
    float4 wq = wqp[fi];
    float4 qv = qp[fi];
    qdot += qv.x * wq.x + qv.y * wq.y + qv.z * wq.z + qv.w * wq.w;
  }
  #pragma unroll
  for (int off = 16; off >= 1; off >>= 1) qdot += __shfl_xor(qdot, off, 32);
  qdot += b_att[0];

  const int nbase = chunk * 128 + wave * 16;
  for (int i = 0; i < 16; ++i) {
    const int n = nbase + i;
    const float4* kp =
        reinterpret_cast<const float4*>(K + ((size_t)b * N_ + n) * H_);
    float s = 0.0f;
    #pragma unroll
    for (int j = 0; j < 4; ++j) {
      float4 kv = kp[lane + 32 * j];
      s += kv.x * wk[j].x + kv.y * wk[j].y + kv.z * wk[j].z + kv.w * wk[j].w;
    }
    #pragma unroll
    for (int off = 16; off >= 1; off >>= 1) s += __shfl_xor(s, off, 32);
    if (lane == 0) {
      const float m = (1.0f - (float)adj[b * N_ + n]) * MASK_NEG;
      alpha[b * N_ + n] = s + qdot - m;
    }
  }
}

// ---------------------------------------------------------------------------
// Kernel B: row softmax over n (one block per batch).
// ---------------------------------------------------------------------------
__global__ void __launch_bounds__(256) softmax_kernel(
    const float* __restrict__ alpha, float* __restrict__ attn)
{
  const int b   = blockIdx.x;
  const int tid = threadIdx.x;
  __shared__ float red[256];
  const float* a = alpha + b * N_;

  float m = -3.4e38f;
  for (int n = tid; n < N_; n += 256) m = fmaxf(m, a[n]);
  red[tid] = m;
  __syncthreads();
  for (int s = 128; s > 0; s >>= 1) {
    if (tid < s) red[tid] = fmaxf(red[tid], red[tid + s]);
    __syncthreads();
  }
  m = red[0];
  __syncthreads();

  float sum = 0.0f;
  for (int n = tid; n < N_; n += 256) sum += __expf(a[n] - m);
  red[tid] = sum;
  __syncthreads();
  for (int s = 128; s > 0; s >>= 1) {
    if (tid < s) red[tid] += red[tid + s];
    __syncthreads();
  }
  const float inv = 1.0f / red[0];
  for (int n = tid; n < N_; n += 256) attn[b * N_ + n] = __expf(a[n] - m) * inv;
}

// ---------------------------------------------------------------------------
// Kernel C: partial weighted sums of V.
//   part[b][chunk][0][h] = sum_{n in chunk} attn*sm*V[b,n,h]
//   part[b][chunk][1][h] = sum_{n in chunk} attn*(1-sm)*V[b,n,h]
// Thread t owns the float2 at h = 2t  -> one b64 load per row per thread,
// 256 lanes x 8B = full 2KB row, perfectly coalesced.
// ---------------------------------------------------------------------------
__global__ void __launch_bounds__(256) wsum_kernel(
    const float* __restrict__ V, const float* __restrict__ attn,
    const int* __restrict__ s_mask, float* __restrict__ part)
{
  const int b = blockIdx.y, chunk = blockIdx.x, tid = threadIdx.x;
  __shared__ float w0s[128], w1s[128];
  const int nbase = chunk * 128;
  if (tid < 128) {
    const float w = attn[b * N_ + nbase + tid];
    const float s = (float)s_mask[b * N_ + nbase + tid];
    w0s[tid] = w * s;
    w1s[tid] = w * (1.0f - s);
  }
  __syncthreads();

  float2 acc0 = {0.f, 0.f}, acc1 = {0.f, 0.f};
  const float2* vp = reinterpret_cast<const float2*>(
      V + ((size_t)b * N_ + nbase) * H_);
  const int hv = tid;                       // float2 index; h = 2*tid
  for (int i = 0; i < 128; ++i) {
    const float2 v = vp[(size_t)i * (H_ / 2) + hv];
    const float w0 = w0s[i], w1 = w1s[i];
    acc0.x += w0 * v.x; acc0.y += w0 * v.y;
    acc1.x += w1 * v.x; acc1.y += w1 * v.y;
  }
  float2* p = reinterpret_cast<float2*>(
      part + (size_t)(b * NCHUNK + chunk) * 2 * H_);
  p[hv] = acc0;
  p[(H_ / 2) + hv] = acc1;
}

// ---------------------------------------------------------------------------
// Kernel D: fold the NCHUNK partials -> u[b][sel][h]   (so that
// u[b*1024 + k] == concat(u0,u1) along k, exactly the GEMM A operand).
// ---------------------------------------------------------------------------
__global__ void __launch_bounds__(256) reduce_kernel(
    const float* __restrict__ part, float* __restrict__ u)
{
  const int idx = blockIdx.x * 256 + threadIdx.x;   // 0 .. B*2*H-1
  const int h   = idx & (H_ - 1);
  const int sel = (idx >> 9) & 1;
  const int b   = idx >> 10;
  float s = 0.0f;
  #pragma unroll
  for (int c = 0; c < NCHUNK; ++c)
    s += part[((size_t)(b * NCHUNK + c) * 2 + sel) * H_ + h];
  u[idx] = s;
}

// ---------------------------------------------------------------------------
// Kernel E: attn_sum = u0*Wr0^T + u1*Wr1^T + Q*Wri^T via V_WMMA_F32_16X16X4_F32
// A-frag (16x4 f32): lanes 0-15 hold M=0..15 with K=0(v0),1(v1);
//                    lanes 16-31 hold K=2(v0),3(v1).
// B-frag (4x16):     v0 = row K=0 (lanes 0-15) / K=2 (lanes 16-31), v1 = K+1.
// C/D (16x16 f32):   VGPR j, lanes 0-15 -> M=j; lanes 16-31 -> M=j+8.
// One wave per 16-wide N tile; both M tiles (rows 0-15, 16-31) share B frags.
// EXEC stays all-ones: no divergent control flow anywhere in this kernel.
// ---------------------------------------------------------------------------
__device__ __forceinline__ void gemm_seg(
    const float* __restrict__ Ap, int lda, const float* __restrict__ Wp,
    int idx, int half, int o0, v8f& c0, v8f& c1)
{
  for (int k = 0; k < H_; k += 4) {
    const int ka = k + 2 * half;
    v2f a0, a1, bf;
    a0.x = Ap[idx * lda + ka];
    a0.y = Ap[idx * lda + ka + 1];
    a1.x = Ap[(16 + idx) * lda + ka];
    a1.y = Ap[(16 + idx) * lda + ka + 1];
    bf.x = Wp[(o0 + idx) * H_ + ka];       // B'(k,o) = W[o,k]
    bf.y = Wp[(o0 + idx) * H_ + ka + 1];
    c0 = __builtin_amdgcn_wmma_f32_16x16x4_f32(false, a0, false, bf,
                                               (short)0, c0, false, false);
    c1 = __builtin_amdgcn_wmma_f32_16x16x4_f32(false, a1, false, bf,
                                               (short)0, c1, false, false);
  }
}

__global__ void __launch_bounds__(256) out_gemm_kernel(
    const float* __restrict__ u, const float* __restrict__ Q,
    const float* __restrict__ Wr0, const float* __restrict__ Wr1,
    const float* __restrict__ Wri, float* __restrict__ attn_sum)
{
  const int tid  = threadIdx.x;
  const int wave = tid >> 5;
  const int lane = tid & 31;
  const int half = lane >> 4;
  const int idx  = lane & 15;
  const int o0   = (blockIdx.x * 8 + wave) * 16;   // 32 N-tiles total

  v8f c0 = {};
  v8f c1 = {};
  gemm_seg(u,        2 * H_, Wr0, idx, half, o0, c0, c1);  // k in [0,512):    u0
  gemm_seg(u + H_,   2 * H_, Wr1, idx, half, o0, c0, c1);  // k in [512,1024): u1
  gemm_seg(Q,        H_,     Wri, idx, half, o0, c0, c1);  // k in [1024,1536): Q

  #pragma unroll
  for (int j = 0; j < 8; ++j) {
    const int row = j + 8 * half;
    attn_sum[row * H_ + o0 + idx]        = c0[j];
    attn_sum[(16 + row) * H_ + o0 + idx] = c1[j];
  }
}

// ---------------------------------------------------------------------------
extern "C" void kernel_launch(void* const* d_in, const int* in_sizes, int n_in,
                              void* d_out, int out_size, void* d_ws, size_t ws_size,
                              hipStream_t stream) {
  const float* Q      = (const float*)d_in[0];
  const float* K      = (const float*)d_in[1];
  const float* V      = (const float*)d_in[2];
  const int*   adj    = (const int*)  d_in[3];
  const int*   s_mask = (const int*)  d_in[4];
  const float* w_att  = (const float*)d_in[5];
  const float* b_att  = (const float*)d_in[6];
  const float* Wr0    = (const float*)d_in[7];
  const float* Wr1    = (const float*)d_in[8];
  const float* Wri    = (const float*)d_in[9];

  float* out      = (float*)d_out;
  float* attn     = out;               // (B,1,N) -> B*N floats
  float* attn_sum = out + B_ * N_;     // (B,H)   -> B*H floats

  float* ws    = (float*)d_ws;
  float* alpha = ws;                                   // B*N
  float* part  = alpha + B_ * N_;                      // B*NCHUNK*2*H
  float* u     = part + (size_t)B_ * NCHUNK * 2 * H_;  // B*2*H

  alpha_kernel  <<<dim3(N_ / 128, B_), 256, 0, stream>>>(Q, K, adj, w_att, b_att, alpha);
  softmax_kernel<<<B_, 256, 0, stream>>>(alpha, attn);
  wsum_kernel   <<<dim3(NCHUNK, B_), 256, 0, stream>>>(V, attn, s_mask, part);
  reduce_kernel <<<(B_ * 2 * H_) / 256, 256, 0, stream>>>(part, u);
  out_gemm_kernel<<<(H_ / 16) / 8, 256, 0, stream>>>(u, Q, Wr0, Wr1, Wri, attn_sum);
}